// Codec_49520972923476
// MI455X (gfx1250) — compile-verified
//
#include <hip/hip_runtime.h>

// Elementwise codec recurrence (f == p in the reference scan, so y_t = f_{t+1}):
//   f0 = 0;  for t in [0,T): g = f - x; spike_t = 0.5*(1+g); f -= lr_t*g; y_t = f
//   out layout: [2, T, N] = [spike planes | y planes], lr_t = 0.15 * 0.95^t
//
// Roofline: 512 MB NT-stored + 16 MB read, ~0.4 FLOP/byte -> pure HBM bound
// (~23 us at 23.3 TB/s). No matrix structure, so no WMMA; optimization is
// b128 non-temporal stores (output >> 192 MB L2, never re-read) with 32-bit
// per-lane offsets so addressing stays in SALU/saddr form.

typedef float v4f __attribute__((ext_vector_type(4)));

__global__ __launch_bounds__(256) void codec_vec4_kernel(
    const float* __restrict__ x, float* __restrict__ out,
    unsigned N, int T) {
  const unsigned base = 4u * (blockIdx.x * blockDim.x + threadIdx.x);
  if (base >= N) return;

  const float* spike_p = out + base;            // plane 0, this lane
  const float* y_p     = out + (unsigned long long)T * N + base;

  v4f xv = *(const v4f*)(x + base);
  v4f f  = {0.f, 0.f, 0.f, 0.f};
  float lr = 0.15f;

#pragma unroll 4
  for (int t = 0; t < T; ++t) {
    v4f g     = f - xv;              // gradient
    v4f spike = 0.5f + 0.5f * g;     // 0.5*(1+g)  -> v_fma
    f         = f - lr * g;          // SGD step; y_t == updated f
    __builtin_nontemporal_store(spike, (v4f*)spike_p);
    __builtin_nontemporal_store(f,     (v4f*)y_p);
    spike_p += N;                    // uniform plane stride -> SALU
    y_p     += N;
    lr *= 0.95f;
  }
}

// Fallback for N not a multiple of 4 (never taken for 2048x2048).
__global__ __launch_bounds__(256) void codec_scalar_kernel(
    const float* __restrict__ x, float* __restrict__ out,
    unsigned N, int T) {
  const unsigned i = blockIdx.x * blockDim.x + threadIdx.x;
  if (i >= N) return;

  const float* spike_p = out + i;
  const float* y_p     = out + (unsigned long long)T * N + i;

  float xv = x[i];
  float f  = 0.f;
  float lr = 0.15f;
  for (int t = 0; t < T; ++t) {
    float g     = f - xv;
    float spike = 0.5f + 0.5f * g;
    f           = f - lr * g;
    __builtin_nontemporal_store(spike, (float*)spike_p);
    __builtin_nontemporal_store(f,     (float*)y_p);
    spike_p += N;
    y_p     += N;
    lr *= 0.95f;
  }
}

extern "C" void kernel_launch(void* const* d_in, const int* in_sizes, int n_in,
                              void* d_out, int out_size, void* d_ws, size_t ws_size,
                              hipStream_t stream) {
  const float* x = (const float*)d_in[0];
  float* out = (float*)d_out;

  const unsigned N = (unsigned)in_sizes[0];   // 2048*2048 elements of x
  const int T = out_size / (int)(2 * N);      // out is [2, T, N]

  const int threads = 256;                    // 8 wave32s / block
  if (N % 4u == 0u) {
    const unsigned nvec = N / 4u;             // one thread per float4
    const int blocks = (int)((nvec + threads - 1) / threads);
    codec_vec4_kernel<<<blocks, threads, 0, stream>>>(x, out, N, T);
  } else {
    const int blocks = (int)((N + threads - 1) / threads);
    codec_scalar_kernel<<<blocks, threads, 0, stream>>>(x, out, N, T);
  }
}